// BipartiteHeteroGNN_67886253081080
// MI455X (gfx1250) — compile-verified
//
#include <hip/hip_runtime.h>
#include <hip/hip_bf16.h>

typedef float v2f __attribute__((ext_vector_type(2)));
typedef float v8f __attribute__((ext_vector_type(8)));

#define HID 64
#define N_CONS 50000
#define N_VALS 100000
#define E_CV 1000000
#define E_VC 1000000
#define E_VV 500000

// --- scalar -> 64 encoder layer: out[i][j] = relu(f(x[i]) * W[j] + b[j]) ---
__global__ __launch_bounds__(256) void k_scalar_encode(
    const float* __restrict__ x, const float* __restrict__ W,
    const float* __restrict__ b, float* __restrict__ out, int n, int mode)
{
    int t = blockIdx.x * blockDim.x + threadIdx.x;
    if (t >= n * HID) return;
    int i = t >> 6, j = t & 63;
    float v = x[i];
    if (mode) v = log1pf(v);
    out[t] = fmaxf(fmaf(v, W[j], b[j]), 0.0f);
}

// --- float4 copy (agg init: agg = x_dst) ---
__global__ __launch_bounds__(256) void k_copy4(
    const float4* __restrict__ s, float4* __restrict__ d, int n4)
{
    int t = blockIdx.x * blockDim.x + threadIdx.x;
    if (t < n4) d[t] = s[t];
}

// --- OUT[n,64] = act(A[n,64] @ W[64,64] + bias) (+= OUT if accum) via f32 WMMA ---
// W staged once per block into LDS in fragment order:
//   fragment (ks,t), lane (half,m) -> WL[(ks*4+t)*64 + half*32 + m*2 + {0,1}]
//     = W[(4ks+2half)+{0,1}][16t+m]
// so each B fragment is one conflict-free ds_load_b64 (256 contiguous LDS bytes/wave).
// A frag (16x4): lane&15 = M, K = 4ks + 2*(lane>>4) + vgpr  (one global b64)
// D (16x16):     lane&15 = N, M = vgpr + 8*(lane>>4)
__global__ __launch_bounds__(256) void k_dense64_wmma(
    const float* __restrict__ A, const float* __restrict__ W,
    const float* __restrict__ bias, float* __restrict__ OUT,
    int n_rows, int do_relu, int do_accum)
{
    __shared__ float WL[HID * HID];   // 16 KB of the WGP's 320 KB LDS
    __shared__ float BL[HID];
    int tid = threadIdx.x;
#pragma unroll
    for (int it = 0; it < 8; ++it) {
        int idx  = tid + 256 * it;           // 2048 fragment-halves
        int ks   = idx >> 7;
        int t    = (idx >> 5) & 3;
        int half = (idx >> 4) & 1;
        int m    = idx & 15;
        int ka   = 4 * ks + 2 * half;
        int col  = 16 * t + m;
        WL[idx * 2 + 0] = W[(size_t)ka * HID + col];
        WL[idx * 2 + 1] = W[(size_t)(ka + 1) * HID + col];
    }
    if (tid < HID) BL[tid] = bias[tid];
    __syncthreads();                          // all threads reach barrier

    int gwave = (int)((blockIdx.x * (unsigned)blockDim.x + tid) >> 5);
    int row0  = gwave * 16;
    if (row0 < n_rows) {                      // uniform per wave: EXEC all-ones
        int lane    = tid & 31;
        int half    = lane >> 4;
        int m       = lane & 15;
        int laneoff = half * 32 + m * 2;

        v8f acc[4] = {};
        const float* Arow = A + (size_t)(row0 + m) * HID;
#pragma unroll
        for (int ks = 0; ks < 16; ++ks) {
            v2f a = *(const v2f*)(Arow + 4 * ks + 2 * half);
#pragma unroll
            for (int t = 0; t < 4; ++t) {
                v2f bfrag = *(const v2f*)(WL + (ks * 4 + t) * 64 + laneoff);
                acc[t] = __builtin_amdgcn_wmma_f32_16x16x4_f32(
                    false, a, false, bfrag, (short)0, acc[t], false, false);
            }
        }
#pragma unroll
        for (int t = 0; t < 4; ++t) {
            int col  = t * 16 + m;
            float bv = BL[col];
#pragma unroll
            for (int v = 0; v < 8; ++v) {
                int row = row0 + v + 8 * half;
                float val = acc[t][v] + bv;
                if (do_relu) val = fmaxf(val, 0.0f);
                size_t idx = (size_t)row * HID + col;
                if (do_accum) val += OUT[idx];
                OUT[idx] = val;
            }
        }
    }
}

// --- edge messages: agg[dst] += relu(X[src] + ea*We + be); 4 threads/edge ---
__global__ __launch_bounds__(256) void k_edge_scatter(
    const float* __restrict__ X, const int* __restrict__ src,
    const int* __restrict__ dst, const float* __restrict__ ea,
    const float* __restrict__ We, const float* __restrict__ be,
    float* __restrict__ agg, int E)
{
    int t = blockIdx.x * blockDim.x + threadIdx.x;
    int e = t >> 2;
    if (e >= E) return;
    int j = (t & 3) * 16;
    float a = ea[e];
    const float* xs = X + (size_t)src[e] * HID + j;
    float* ad = agg + (size_t)dst[e] * HID + j;
#pragma unroll
    for (int jj = 0; jj < 16; jj += 4) {
        float4 x  = *(const float4*)(xs + jj);
        float4 w  = *(const float4*)(We + j + jj);
        float4 bb = *(const float4*)(be + j + jj);
        atomicAdd(ad + jj + 0, fmaxf(x.x + fmaf(a, w.x, bb.x), 0.0f));
        atomicAdd(ad + jj + 1, fmaxf(x.y + fmaf(a, w.y, bb.y), 0.0f));
        atomicAdd(ad + jj + 2, fmaxf(x.z + fmaf(a, w.z, bb.z), 0.0f));
        atomicAdd(ad + jj + 3, fmaxf(x.w + fmaf(a, w.w, bb.w), 0.0f));
    }
}

// --- prediction head final layer: out[i] = dot(H[i,:], w) + b ---
__global__ __launch_bounds__(256) void k_pred_dot(
    const float* __restrict__ H, const float* __restrict__ w,
    const float* __restrict__ b, float* __restrict__ out, int n)
{
    int i = blockIdx.x * blockDim.x + threadIdx.x;
    if (i >= n) return;
    const float* h = H + (size_t)i * HID;
    float s = 0.0f;
#pragma unroll
    for (int j = 0; j < HID; ++j) s = fmaf(h[j], w[j], s);
    out[i] = s + b[0];
}

static inline unsigned gdiv(long n, long d) { return (unsigned)((n + d - 1) / d); }

extern "C" void kernel_launch(void* const* d_in, const int* in_sizes, int n_in,
                              void* d_out, int out_size, void* d_ws, size_t ws_size,
                              hipStream_t stream)
{
    (void)in_sizes; (void)n_in; (void)out_size; (void)ws_size;
    const float* b_in   = (const float*)d_in[0];
    const float* q_in   = (const float*)d_in[1];
    const float* xs_in  = (const float*)d_in[2];
    const int* c2v_src  = (const int*)d_in[3];
    const int* c2v_dst  = (const int*)d_in[4];
    const int* v2c_src  = (const int*)d_in[5];
    const int* v2c_dst  = (const int*)d_in[6];
    const int* v2v_src  = (const int*)d_in[7];
    const int* v2v_dst  = (const int*)d_in[8];
    const float* ea_c2v = (const float*)d_in[9];
    const float* ea_v2c = (const float*)d_in[10];
    const float* ea_v2v = (const float*)d_in[11];

    // params flattened (jax pytree: dict keys alphabetical):
    // 12..15 b_enc; 16+18*l: layer l {c2v:0..5, v2c:6..11, v2v:12..17}
    //   each conv = edge.W, edge.b, mlp0.W, mlp0.b, mlp1.W, mlp1.b
    // 52..55 pred; 56..59 q_enc; 60..63 x_enc
    const float* P[64];
    for (int i = 12; i < 64; ++i) P[i] = (const float*)d_in[i];

    const size_t CONS = (size_t)N_CONS * HID;
    const size_t VALS = (size_t)N_VALS * HID;
    float* ws    = (float*)d_ws;
    float* cons  = ws;
    float* consN = cons  + CONS;
    float* aggc  = consN + CONS;
    float* vals  = aggc  + CONS;
    float* valsN = vals  + VALS;
    float* aggv  = valsN + VALS;

    auto dense = [&](const float* A, const float* Wp, const float* bp, float* O,
                     int n, int relu, int accum) {
        int waves = n / 16;
        k_dense64_wmma<<<gdiv(waves, 8), 256, 0, stream>>>(A, Wp, bp, O, n, relu, accum);
    };
    auto copyf = [&](const float* s, float* d, size_t nf) {
        k_copy4<<<gdiv((long)(nf / 4), 256), 256, 0, stream>>>(
            (const float4*)s, (float4*)d, (int)(nf / 4));
    };

    // ---- encoders ----
    k_scalar_encode<<<gdiv((long)N_CONS * HID, 256), 256, 0, stream>>>(
        b_in, P[12], P[13], aggc, N_CONS, 0);
    dense(aggc, P[14], P[15], cons, N_CONS, 0, 0);

    k_scalar_encode<<<gdiv((long)N_VALS * HID, 256), 256, 0, stream>>>(
        xs_in, P[60], P[61], aggv, N_VALS, 1);                 // log1p path
    dense(aggv, P[62], P[63], vals, N_VALS, 0, 0);
    k_scalar_encode<<<gdiv((long)N_VALS * HID, 256), 256, 0, stream>>>(
        q_in, P[56], P[57], aggv, N_VALS, 0);
    dense(aggv, P[58], P[59], vals, N_VALS, 0, 1);             // vals = x_enc + q_enc

    // ---- GNN layers ----
    for (int l = 0; l < 2; ++l) {
        int B = 16 + 18 * l;
        // v2c: cons_new = mlp(cons + sum_msgs(vals))
        copyf(cons, aggc, CONS);
        k_edge_scatter<<<gdiv((long)E_VC * 4, 256), 256, 0, stream>>>(
            vals, v2c_src, v2c_dst, ea_v2c, P[B + 6], P[B + 7], aggc, E_VC);
        dense(aggc, P[B + 8],  P[B + 9],  aggc,  N_CONS, 1, 0);
        dense(aggc, P[B + 10], P[B + 11], consN, N_CONS, 0, 0);
        // c2v: vals_c = mlp(vals + sum_msgs(cons_new))
        copyf(vals, aggv, VALS);
        k_edge_scatter<<<gdiv((long)E_CV * 4, 256), 256, 0, stream>>>(
            consN, c2v_src, c2v_dst, ea_c2v, P[B + 0], P[B + 1], aggv, E_CV);
        dense(aggv, P[B + 2], P[B + 3], aggv,  N_VALS, 1, 0);
        dense(aggv, P[B + 4], P[B + 5], valsN, N_VALS, 0, 0);
        // v2v: vals = vals_c + mlp(vals + sum_msgs(vals))
        copyf(vals, aggv, VALS);
        k_edge_scatter<<<gdiv((long)E_VV * 4, 256), 256, 0, stream>>>(
            vals, v2v_src, v2v_dst, ea_v2v, P[B + 12], P[B + 13], aggv, E_VV);
        dense(aggv, P[B + 14], P[B + 15], aggv,  N_VALS, 1, 0);
        dense(aggv, P[B + 16], P[B + 17], valsN, N_VALS, 0, 1); // += vals_c
        // swap ping-pong
        float* tc = cons; cons = consN; consN = tc;
        float* tv = vals; vals = valsN; valsN = tv;
    }

    // ---- prediction head ----
    dense(vals, P[52], P[53], aggv, N_VALS, 1, 0);
    k_pred_dot<<<gdiv((long)N_VALS, 256), 256, 0, stream>>>(
        aggv, P[54], P[55], (float*)d_out, N_VALS);
}